// gcn_encoder_60842506715396
// MI455X (gfx1250) — compile-verified
//
#include <hip/hip_runtime.h>
#include <cmath>

#define GHEADS 3
#define GATT   64

typedef float v2f __attribute__((ext_vector_type(2)));
typedef float v8f __attribute__((ext_vector_type(8)));

// ---------------------------------------------------------------------------
// Dense GEMM  Y[n,K] = X[n,M] @ W[M,K]  via V_WMMA_F32_16X16X4_F32.
// One wave (32 threads) computes a 16x64 output strip: 4 column tiles share
// one A fragment per k-step (A reuse x4, address math amortized).
// ISA layouts (CDNA5 ISA §7.12.2):
//   A 16x4 : lanes 0-15 -> {K=0,K=1}, lanes 16-31 -> {K=2,K=3}, M = lane&15
//   B 4x16 : vgpr0 -> K=0(l<16)/K=2(l>=16), vgpr1 -> K=1/K=3, N = lane&15
//   C/D    : vgpr v -> M=v (l<16) / M=v+8 (l>=16), N = lane&15
// Requires Kdim % 64 == 0 (true for 128 and 192).
// ---------------------------------------------------------------------------
__global__ void wmma_gemm_f32(const float* __restrict__ X,
                              const float* __restrict__ W,
                              float* __restrict__ Y,
                              int n_rows, int Mdim, int Kdim) {
    int lane = threadIdx.x & 31;
    int col_base = blockIdx.x * 64;     // 4 column tiles of 16
    int row_base = blockIdx.y * 16;
    int half = lane >> 4;               // 0: lanes 0-15, 1: lanes 16-31
    int l    = lane & 15;

    int arow = row_base + l;
    if (arow >= n_rows) arow = n_rows - 1;      // clamp (rows guarded at store)

    const float* xp = X + (size_t)arow * Mdim + 2 * half;
    const float* wp = W + (size_t)(2 * half) * Kdim + col_base + l;

    v8f acc[4];
#pragma unroll
    for (int j = 0; j < 4; ++j) acc[j] = (v8f){0.f,0.f,0.f,0.f,0.f,0.f,0.f,0.f};

    for (int k = 0; k < Mdim; k += 4) {
        v2f a;
        a.x = xp[k];
        a.y = xp[k + 1];
        const float* wk0 = wp + (size_t)k * Kdim;
        const float* wk1 = wp + (size_t)(k + 1) * Kdim;
#pragma unroll
        for (int j = 0; j < 4; ++j) {
            v2f b;
            b.x = wk0[j * 16];
            b.y = wk1[j * 16];
            acc[j] = __builtin_amdgcn_wmma_f32_16x16x4_f32(
                /*neg_a=*/false, a, /*neg_b=*/false, b,
                /*c_mod=*/(short)0, acc[j],
                /*reuse_a=*/false, /*reuse_b=*/false);
        }
    }

#pragma unroll
    for (int j = 0; j < 4; ++j) {
#pragma unroll
        for (int v = 0; v < 8; ++v) {
            int row = row_base + v + 8 * half;
            if (row < n_rows)
                Y[(size_t)row * Kdim + col_base + j * 16 + l] = acc[j][v];
        }
    }
}

// ---------------------------------------------------------------------------
// Small utility kernels
// ---------------------------------------------------------------------------
__global__ void fill_f32(float* p, float v, int n) {
    int i = blockIdx.x * blockDim.x + threadIdx.x;
    if (i < n) p[i] = v;
}

// monotonic float -> uint encoding for atomicMax-based segment max
__device__ __forceinline__ unsigned enc_f(float f) {
    unsigned u = __float_as_uint(f);
    return (u & 0x80000000u) ? ~u : (u | 0x80000000u);
}
__device__ __forceinline__ float dec_f(unsigned s) {
    return (s & 0x80000000u) ? __uint_as_float(s ^ 0x80000000u)
                             : __uint_as_float(~s);
}

__global__ void fill_enc_neginf(unsigned* p, int n) {
    int i = blockIdx.x * blockDim.x + threadIdx.x;
    if (i < n) p[i] = enc_f(-1e30f);
}

__global__ void bias_rows(float* __restrict__ out, const float* __restrict__ b,
                          int n_nodes, int K) {
    int i = blockIdx.x * blockDim.x + threadIdx.x;
    if (i < n_nodes * K) out[i] = b[i % K];
}

// ---------------------------------------------------------------------------
// Degree (dst side, self-loops included) and deg^-1/2
// ---------------------------------------------------------------------------
__global__ void deg_count(const int* __restrict__ edges, int E, int Nn,
                          float* __restrict__ deg) {
    int e = blockIdx.x * blockDim.x + threadIdx.x;
    int total = E + Nn;
    if (e >= total) return;
    int dst = (e < E) ? edges[E + e] : (e - E);
    atomicAdd(&deg[dst], 1.0f);
}

__global__ void deg_inv(const float* __restrict__ deg, float* __restrict__ dinv,
                        int Nn) {
    int i = blockIdx.x * blockDim.x + threadIdx.x;
    if (i >= Nn) return;
    float d = deg[i];
    dinv[i] = (d > 0.f) ? rsqrtf(d) : 0.f;
}

// ---------------------------------------------------------------------------
// GCN scatter: out[dst] += h[src] * dinv[src]*dinv[dst].  One block per edge.
// ---------------------------------------------------------------------------
__global__ void gcn_scatter(const float* __restrict__ h,
                            const float* __restrict__ dinv,
                            const int* __restrict__ edges, int E, int Nn,
                            float* __restrict__ out, int K) {
    int e = blockIdx.x;
    int src, dst;
    if (e < E) { src = edges[e]; dst = edges[E + e]; }
    else       { src = dst = e - E; }
    float nrm = dinv[src] * dinv[dst];
    int t = threadIdx.x;
    atomicAdd(&out[(size_t)dst * K + t], h[(size_t)src * K + t] * nrm);
}

// ---------------------------------------------------------------------------
// LayerNorm + Mish (in place). One wave32 per node; K in {128,192}.
// ---------------------------------------------------------------------------
__global__ void ln_mish(float* __restrict__ x, const float* __restrict__ g,
                        const float* __restrict__ b, int Nn, int K) {
    int wave = threadIdx.x >> 5;
    int lane = threadIdx.x & 31;
    int node = blockIdx.x * (blockDim.x >> 5) + wave;
    if (node >= Nn) return;                 // whole wave exits together

    float* xp = x + (size_t)node * K;
    int per = K >> 5;                       // 4 or 6
    float vals[6];
    float s = 0.f;
    for (int i = 0; i < per; ++i) { vals[i] = xp[lane + i * 32]; s += vals[i]; }
    for (int o = 16; o >= 1; o >>= 1) s += __shfl_xor(s, o, 32);
    float mean = s / (float)K;

    float s2 = 0.f;
    for (int i = 0; i < per; ++i) { float d = vals[i] - mean; s2 += d * d; }
    for (int o = 16; o >= 1; o >>= 1) s2 += __shfl_xor(s2, o, 32);
    float rstd = rsqrtf(s2 / (float)K + 1e-5f);

    for (int i = 0; i < per; ++i) {
        int c = lane + i * 32;
        float y = (vals[i] - mean) * rstd * g[c] + b[c];
        float sp = (y > 20.f) ? y : log1pf(expf(y));   // softplus
        xp[c] = y * tanhf(sp);                          // mish
    }
}

// ---------------------------------------------------------------------------
// GAT attention
// ---------------------------------------------------------------------------
__global__ void gat_alpha(const float* __restrict__ h2,
                          const float* __restrict__ a_s,
                          const float* __restrict__ a_d,
                          float* __restrict__ as_, float* __restrict__ ad_,
                          int Nn) {
    int idx = blockIdx.x * blockDim.x + threadIdx.x;
    if (idx >= Nn * GHEADS) return;
    int node = idx / GHEADS;
    int head = idx - node * GHEADS;
    const float* hp = h2 + (size_t)node * (GHEADS * GATT) + head * GATT;
    float s = 0.f, d = 0.f;
    for (int c = 0; c < GATT; ++c) {
        float v = hp[c];
        s += v * a_s[head * GATT + c];
        d += v * a_d[head * GATT + c];
    }
    as_[idx] = s;
    ad_[idx] = d;
}

__global__ void edge_max(const int* __restrict__ edges, int E, int Nn,
                         const float* __restrict__ as_,
                         const float* __restrict__ ad_,
                         unsigned* __restrict__ emax) {
    int idx = blockIdx.x * blockDim.x + threadIdx.x;
    int total = (E + Nn) * GHEADS;
    if (idx >= total) return;
    int e = idx / GHEADS;
    int head = idx - e * GHEADS;
    int src, dst;
    if (e < E) { src = edges[e]; dst = edges[E + e]; }
    else       { src = dst = e - E; }
    float v = as_[src * GHEADS + head] + ad_[dst * GHEADS + head];
    v = (v > 0.f) ? v : 0.2f * v;           // leaky relu
    atomicMax(&emax[dst * GHEADS + head], enc_f(v));
}

__global__ void edge_exp(const int* __restrict__ edges, int E, int Nn,
                         const float* __restrict__ as_,
                         const float* __restrict__ ad_,
                         const unsigned* __restrict__ emax,
                         float* __restrict__ eattn,
                         float* __restrict__ denom) {
    int idx = blockIdx.x * blockDim.x + threadIdx.x;
    int total = (E + Nn) * GHEADS;
    if (idx >= total) return;
    int e = idx / GHEADS;
    int head = idx - e * GHEADS;
    int src, dst;
    if (e < E) { src = edges[e]; dst = edges[E + e]; }
    else       { src = dst = e - E; }
    float v = as_[src * GHEADS + head] + ad_[dst * GHEADS + head];
    v = (v > 0.f) ? v : 0.2f * v;
    float ex = expf(v - dec_f(emax[dst * GHEADS + head]));
    eattn[idx] = ex;
    atomicAdd(&denom[dst * GHEADS + head], ex);
}

__global__ void gat_scatter(const float* __restrict__ h2,
                            const float* __restrict__ eattn,
                            const float* __restrict__ denom,
                            const int* __restrict__ edges, int E, int Nn,
                            float* __restrict__ out) {
    int e = blockIdx.x;
    int src, dst;
    if (e < E) { src = edges[e]; dst = edges[E + e]; }
    else       { src = dst = e - E; }
    int t = threadIdx.x;                    // 0..191
    int head = t >> 6;                      // GATT = 64
    float w = eattn[(size_t)e * GHEADS + head] /
              (denom[dst * GHEADS + head] + 1e-16f);
    atomicAdd(&out[(size_t)dst * (GHEADS * GATT) + t],
              h2[(size_t)src * (GHEADS * GATT) + t] * w);
}

// ---------------------------------------------------------------------------
// Orchestration
// ---------------------------------------------------------------------------
extern "C" void kernel_launch(void* const* d_in, const int* in_sizes, int n_in,
                              void* d_out, int out_size, void* d_ws,
                              size_t ws_size, hipStream_t stream) {
    (void)n_in; (void)out_size; (void)ws_size;

    const float* x     = (const float*)d_in[0];
    const int*   edges = (const int*)  d_in[1];
    const float* W1    = (const float*)d_in[2];
    const float* b1    = (const float*)d_in[3];
    const float* g1    = (const float*)d_in[4];
    const float* be1   = (const float*)d_in[5];
    const float* Wg    = (const float*)d_in[6];
    const float* a_src = (const float*)d_in[7];
    const float* a_dst = (const float*)d_in[8];
    const float* bg    = (const float*)d_in[9];
    const float* g2    = (const float*)d_in[10];
    const float* be2   = (const float*)d_in[11];
    const float* W2    = (const float*)d_in[12];
    const float* b2    = (const float*)d_in[13];

    const int Nn = in_sizes[0] / 128;   // 50000
    const int E  = in_sizes[1] / 2;     // 400000
    const int TE = E + Nn;              // edges incl. self loops

    float* out1 = (float*)d_out;                 // x1 after LN+mish  [N,128]
    float* out2 = out1 + (size_t)Nn * 128;       // final gcn result  [N,128]

    // workspace layout (floats)
    float*    bufA  = (float*)d_ws;                       // [N,192] h1 / h3
    float*    bufB  = bufA + (size_t)Nn * 192;            // [N,192] h2 (GAT lin)
    float*    bufC  = bufB + (size_t)Nn * 192;            // [N,192] x2 accum
    float*    deg   = bufC + (size_t)Nn * 192;            // [N]
    float*    dinv  = deg  + Nn;                          // [N]
    float*    as_   = dinv + Nn;                          // [N,3]
    float*    ad_   = as_  + (size_t)Nn * GHEADS;         // [N,3]
    unsigned* emax  = (unsigned*)(ad_ + (size_t)Nn * GHEADS);   // [N,3] encoded
    float*    denom = (float*)(emax + (size_t)Nn * GHEADS);     // [N,3]
    float*    eattn = denom + (size_t)Nn * GHEADS;        // [TE,3]

    const int tiles_m = (Nn + 15) / 16;
    dim3 wave(32);

    // ---- degrees (shared by both GCN layers) ----
    fill_f32<<<(Nn + 255) / 256, 256, 0, stream>>>(deg, 0.f, Nn);
    deg_count<<<(TE + 255) / 256, 256, 0, stream>>>(edges, E, Nn, deg);
    deg_inv<<<(Nn + 255) / 256, 256, 0, stream>>>(deg, dinv, Nn);

    // ---- GCN layer 1 ----
    wmma_gemm_f32<<<dim3(128 / 64, tiles_m), wave, 0, stream>>>(x, W1, bufA, Nn, 128, 128);
    bias_rows<<<(Nn * 128 + 255) / 256, 256, 0, stream>>>(out1, b1, Nn, 128);
    gcn_scatter<<<TE, 128, 0, stream>>>(bufA, dinv, edges, E, Nn, out1, 128);
    ln_mish<<<(Nn + 7) / 8, 256, 0, stream>>>(out1, g1, be1, Nn, 128);

    // ---- GAT layer ----
    wmma_gemm_f32<<<dim3(192 / 64, tiles_m), wave, 0, stream>>>(out1, Wg, bufB, Nn, 128, 192);
    gat_alpha<<<(Nn * GHEADS + 255) / 256, 256, 0, stream>>>(bufB, a_src, a_dst, as_, ad_, Nn);
    fill_enc_neginf<<<(Nn * GHEADS + 255) / 256, 256, 0, stream>>>(emax, Nn * GHEADS);
    fill_f32<<<(Nn * GHEADS + 255) / 256, 256, 0, stream>>>(denom, 0.f, Nn * GHEADS);
    edge_max<<<(TE * GHEADS + 255) / 256, 256, 0, stream>>>(edges, E, Nn, as_, ad_, emax);
    edge_exp<<<(TE * GHEADS + 255) / 256, 256, 0, stream>>>(edges, E, Nn, as_, ad_, emax, eattn, denom);
    bias_rows<<<(Nn * 192 + 255) / 256, 256, 0, stream>>>(bufC, bg, Nn, 192);
    gat_scatter<<<TE, 192, 0, stream>>>(bufB, eattn, denom, edges, E, Nn, bufC);
    ln_mish<<<(Nn + 7) / 8, 256, 0, stream>>>(bufC, g2, be2, Nn, 192);

    // ---- GCN layer 2 ----
    wmma_gemm_f32<<<dim3(128 / 64, tiles_m), wave, 0, stream>>>(bufC, W2, bufA, Nn, 192, 128);
    bias_rows<<<(Nn * 128 + 255) / 256, 256, 0, stream>>>(out2, b2, Nn, 128);
    gcn_scatter<<<TE, 128, 0, stream>>>(bufA, dinv, edges, E, Nn, out2, 128);
}